// BboxLoss_40003325395671
// MI455X (gfx1250) — compile-verified
//
#include <hip/hip_runtime.h>

#define EPS 1e-7f

typedef __attribute__((ext_vector_type(2))) float v2f;
typedef __attribute__((ext_vector_type(8))) float v8f;

// ===================== helpers =====================

__device__ __forceinline__ unsigned long long u64min(unsigned long long a,
                                                     unsigned long long b) {
  return a < b ? a : b;
}

// wave32 min-reduction of a packed u64 key; xor butterfly -> result in ALL lanes
__device__ __forceinline__ unsigned long long wave_min_u64(unsigned long long v) {
#pragma unroll
  for (int m = 16; m > 0; m >>= 1) {
    unsigned lo = (unsigned)__shfl_xor((int)(unsigned)v, m, 32);
    unsigned hi = (unsigned)__shfl_xor((int)(unsigned)(v >> 32), m, 32);
    v = u64min(v, ((unsigned long long)hi << 32) | lo);
  }
  return v;
}

// total-order float <-> uint mapping (monotone, handles any sign)
__device__ __forceinline__ unsigned f2ord(float f) {
  unsigned b = __float_as_uint(f);
  return b ^ (((int)b < 0) ? 0xFFFFFFFFu : 0x80000000u);
}
__device__ __forceinline__ float ord2f(unsigned u) {
  unsigned b = u ^ (((int)u < 0) ? 0x80000000u : 0xFFFFFFFFu);
  return __uint_as_float(b);
}

// ============ Phase 1: pairwise loss via V_WMMA_F32_16X16X4_F32 ============
// One wave per 16(T) x 16(P) tile. The MSE cross term tgt . pred is a K=4 f32
// GEMM -> exactly the 16x16x4 f32 WMMA shape. IoU clipping done in epilogue.
__global__ __launch_bounds__(256) void pairwise_wmma(
    const float4* __restrict__ pred4, const float4* __restrict__ tgt4,
    float* __restrict__ L, int P, int T) {
  const int tilesP = P >> 4;
  const int tilesT = T >> 4;
  const int wid = threadIdx.x >> 5;
  const int lane = threadIdx.x & 31;
  const long tile = (long)blockIdx.x * 8 + wid;          // wave-uniform
  if (tile >= (long)tilesP * tilesT) return;             // wave-uniform exit
  const int t0 = (int)(tile / tilesP) << 4;
  const int p0 = (int)(tile % tilesP) << 4;
  const int half = lane >> 4;                            // 0: K=0,1  1: K=2,3
  const int lid = lane & 15;

  // A: tgt tile 16x4. lanes 0-15 -> (cx,cy) of row lid; lanes 16-31 -> (w,h)
  const float4 tq = tgt4[t0 + lid];
  v2f a;
  a[0] = half ? tq.z : tq.x;
  a[1] = half ? tq.w : tq.y;
  // B: pred^T 4x16, N along lanes, mirrored K split
  const float4 pq = pred4[p0 + lid];
  v2f b;
  b[0] = half ? pq.z : pq.x;
  b[1] = half ? pq.w : pq.y;

  v8f c = {};
  c = __builtin_amdgcn_wmma_f32_16x16x4_f32(false, a, false, b, (short)0, c,
                                            false, false);
  // c[r] = dot(tgt[t0 + r + 8*half], pred[p0 + lid])

  // per-lane pred scalars (this lane's column is pred p0+lid; pq already held)
  const float x0p = fmaxf(pq.x - 0.5f * pq.z, 0.f);
  const float x1p = fminf(pq.x + 0.5f * pq.z, 1.f);
  const float y0p = fmaxf(pq.y - 0.5f * pq.w, 0.f);
  const float y1p = fminf(pq.y + 0.5f * pq.w, 1.f);
  const float area_p = pq.z * pq.w;
  const float sq_p = pq.x * pq.x + pq.y * pq.y + pq.z * pq.z + pq.w * pq.w;

#pragma unroll
  for (int r = 0; r < 8; ++r) {
    const int t = t0 + r + (half << 3);                  // C/D row layout
    const float4 tr = tgt4[t];                           // L0/L2 broadcast
    const float x0t = tr.x - 0.5f * tr.z, x1t = tr.x + 0.5f * tr.z;
    const float y0t = tr.y - 0.5f * tr.w, y1t = tr.y + 0.5f * tr.w;
    const float ox0 = fmaxf(x0t, x0p), ox1 = fminf(x1t, x1p);
    const float oy0 = fmaxf(y0t, y0p), oy1 = fminf(y1t, y1p);
    const bool no_ov = (ox1 < ox0) || (oy1 < oy0);       // strict, per ref
    const float inter = (ox1 - ox0) * (oy1 - oy0);
    const float area_t = tr.z * tr.w;
    const float iou = inter / (area_p + area_t - inter + EPS);
    const float sq_t = tr.x * tr.x + tr.y * tr.y + tr.z * tr.z + tr.w * tr.w;
    const float mse = (sq_t + sq_p - 2.f * c[r]) * 0.25f;
    L[(size_t)t * P + (p0 + lid)] = no_ov ? (1.f + mse) : (1.f - iou);
  }
}

// ============ Phase 2: sequential greedy argmin over stored L ============
// 1 block x 1024 threads (32 waves). Packed u64 (ordered loss | pred index)
// min gives exact first-occurrence argmin. Availability bits in registers.
// Row t+1 is prefetched into registers while row t is being reduced: the
// loads are mask-independent, only key-building consults the avail bits.
__global__ __launch_bounds__(1024) void greedy_stored(
    const float* __restrict__ L, float* __restrict__ out, int P, int T) {
  const int tid = threadIdx.x;
  const int lane = tid & 31, wid = tid >> 5;
  __shared__ unsigned long long s_wave[32];

  unsigned avail = 0;
#pragma unroll
  for (int k = 0; k < 8; ++k) {
    const int p = tid + (k << 10);
    if (p < P) avail |= 1u << k;
  }

  // preload row 0
  float v[8];
#pragma unroll
  for (int k = 0; k < 8; ++k) {
    const int p = tid + (k << 10);
    v[k] = (p < P) ? L[p] : 0.f;
  }

  float acc = 0.f;
  for (int t = 0; t < T; ++t) {
    // keys for row t from pre-loaded values
    unsigned long long key = ~0ull;
#pragma unroll
    for (int k = 0; k < 8; ++k) {
      if (avail & (1u << k)) {
        const int p = tid + (k << 10);
        key = u64min(key,
                     ((unsigned long long)f2ord(v[k]) << 32) | (unsigned)p);
      }
    }

    // prefetch row t+1 (unconditional wrt avail; latency hides under reduce)
    const int tn = (t + 1 < T) ? (t + 1) : t;
    const float* nrow = L + (size_t)tn * P;
    float nv[8];
#pragma unroll
    for (int k = 0; k < 8; ++k) {
      const int p = tid + (k << 10);
      nv[k] = (p < P) ? nrow[p] : 0.f;
    }

    // two-level argmin: wave butterfly (all lanes hold result), then every
    // wave redundantly reduces the 32 wave partials -> no s_win round-trip
    key = wave_min_u64(key);
    if (lane == 0) s_wave[wid] = key;
    __syncthreads();
    const unsigned long long win = wave_min_u64(s_wave[lane]);

    const int wp = (int)(unsigned)win;
    if ((wp & 1023) == tid) avail &= ~(1u << (wp >> 10));  // remove matched
    if (tid == 0) acc += ord2f((unsigned)(win >> 32));
    __syncthreads();  // protect s_wave before next iteration's writes

#pragma unroll
    for (int k = 0; k < 8; ++k) v[k] = nv[k];
  }
  if (tid == 0) out[0] = acc / (float)T;
}

// ============ Fallback: fused greedy (no workspace), recompute row ============
__global__ __launch_bounds__(1024) void greedy_fused(
    const float4* __restrict__ pred4, const float4* __restrict__ tgt4,
    float* __restrict__ out, int P, int T) {
  const int tid = threadIdx.x;
  const int lane = tid & 31, wid = tid >> 5;
  __shared__ unsigned long long s_wave[32];

  unsigned avail = 0;
#pragma unroll
  for (int k = 0; k < 8; ++k) {
    const int p = tid + (k << 10);
    if (p < P) avail |= 1u << k;
  }

  float acc = 0.f;
  for (int t = 0; t < T; ++t) {
    const float4 tq = tgt4[t];
    const float x0t = tq.x - 0.5f * tq.z, x1t = tq.x + 0.5f * tq.z;
    const float y0t = tq.y - 0.5f * tq.w, y1t = tq.y + 0.5f * tq.w;
    const float area_t = tq.z * tq.w;

    unsigned long long key = ~0ull;
#pragma unroll
    for (int k = 0; k < 8; ++k) {
      if (avail & (1u << k)) {
        const int p = tid + (k << 10);
        const float4 pq = pred4[p];
        const float x0p = fmaxf(pq.x - 0.5f * pq.z, 0.f);
        const float x1p = fminf(pq.x + 0.5f * pq.z, 1.f);
        const float y0p = fmaxf(pq.y - 0.5f * pq.w, 0.f);
        const float y1p = fminf(pq.y + 0.5f * pq.w, 1.f);
        const float ox0 = fmaxf(x0t, x0p), ox1 = fminf(x1t, x1p);
        const float oy0 = fmaxf(y0t, y0p), oy1 = fminf(y1t, y1p);
        const bool no_ov = (ox1 < ox0) || (oy1 < oy0);
        const float inter = (ox1 - ox0) * (oy1 - oy0);
        const float area_p = pq.z * pq.w;
        const float iou = inter / (area_p + area_t - inter + EPS);
        const float dx = pq.x - tq.x, dy = pq.y - tq.y;
        const float dw = pq.z - tq.z, dh = pq.w - tq.w;
        const float mse = (dx * dx + dy * dy + dw * dw + dh * dh) * 0.25f;
        const float vv = no_ov ? (1.f + mse) : (1.f - iou);
        key = u64min(key, ((unsigned long long)f2ord(vv) << 32) | (unsigned)p);
      }
    }
    key = wave_min_u64(key);
    if (lane == 0) s_wave[wid] = key;
    __syncthreads();
    const unsigned long long win = wave_min_u64(s_wave[lane]);
    const int wp = (int)(unsigned)win;
    if ((wp & 1023) == tid) avail &= ~(1u << (wp >> 10));
    if (tid == 0) acc += ord2f((unsigned)(win >> 32));
    __syncthreads();
  }
  if (tid == 0) out[0] = acc / (float)T;
}

// ===================== host launcher =====================
extern "C" void kernel_launch(void* const* d_in, const int* in_sizes, int n_in,
                              void* d_out, int out_size, void* d_ws,
                              size_t ws_size, hipStream_t stream) {
  const float* pred = (const float*)d_in[0];  // [1, P, 4]
  const float* tgt = (const float*)d_in[1];   // [1, T, 4]
  const int P = in_sizes[0] / 4;              // 8192
  const int T = in_sizes[1] / 4;              // 2048
  float* out = (float*)d_out;

  const size_t needL = (size_t)P * (size_t)T * sizeof(float);  // 64 MB
  if (ws_size >= needL && (P % 16) == 0 && (T % 16) == 0 && P <= 8192) {
    // Phase 1: fully parallel, WMMA 16x16x4 f32 per wave tile
    const long tiles = (long)(P >> 4) * (long)(T >> 4);
    const int blocks = (int)((tiles + 7) / 8);
    pairwise_wmma<<<blocks, 256, 0, stream>>>(
        (const float4*)pred, (const float4*)tgt, (float*)d_ws, P, T);
    // Phase 2: sequential greedy over L2-resident L
    greedy_stored<<<1, 1024, 0, stream>>>((const float*)d_ws, out, P, T);
  } else {
    greedy_fused<<<1, 1024, 0, stream>>>((const float4*)pred,
                                         (const float4*)tgt, out, P, T);
  }
}